// GAT_29618094473652
// MI455X (gfx1250) — compile-verified
//
#include <hip/hip_runtime.h>
#include <hip/hip_bf16.h>
#include <math.h>

// Problem constants (match reference)
#define NNODES   50000
#define NEDGES   800000
#define ETOT     (NNODES + NEDGES)   // self-loops appended: 850000
#define DIN      128
#define H1       8
#define C1       16
#define HC1      (H1 * C1)           // 128
#define H2       1
#define C2       16
#define NEG_SLOPE 0.2f

typedef float v2f __attribute__((ext_vector_type(2)));
typedef float v8f __attribute__((ext_vector_type(8)));

// ---------------------------------------------------------------------------
// fill kernel (workspace init each call; d_ws is poisoned, so re-init always)
// ---------------------------------------------------------------------------
__global__ void gat_fill(float* __restrict__ p, float v, long n) {
  long i = (long)blockIdx.x * blockDim.x + threadIdx.x;
  if (i < n) p[i] = v;
}

// ---------------------------------------------------------------------------
// fp32 GEMM on the WMMA pipe: Out[M x NC] = A[M x K] * W[K x NC]
// M % 16 == 0, K % 4 == 0, NC % 16 == 0 (always true here).
// One 16x16 output tile per wave32, K-loop in steps of 4 via
// V_WMMA_F32_16X16X4_F32. Blocks of 8 waves cover one row tile x 8 col
// tiles, so the A rows are reused out of L0/L1 across the block.
// ---------------------------------------------------------------------------
__global__ void gat_wmma_gemm_f32(const float* __restrict__ A,
                                  const float* __restrict__ W,
                                  float* __restrict__ Out,
                                  int M, int K, int NC) {
  const int wave = (blockIdx.x * blockDim.x + threadIdx.x) >> 5;
  const int lane = threadIdx.x & 31;
  const int ntc  = NC >> 4;
  const int ntr  = M >> 4;
  if (wave >= ntr * ntc) return;
  const int rt   = wave / ntc;
  const int ct   = wave % ntc;
  const int row0 = rt << 4;
  const int col0 = ct << 4;
  const int half = lane >> 4;   // 0 -> K pair {0,1}; 1 -> K pair {2,3}
  const int m    = lane & 15;   // row index for A, col index for B/C

  v8f c = {};
  const float* __restrict__ arow = A + (long)(row0 + m) * K;
  for (int k = 0; k < K; k += 4) {
    v2f a, b;
    const int k0 = k + 2 * half;
    a.x = arow[k0 + 0];
    a.y = arow[k0 + 1];
    b.x = W[(long)(k0 + 0) * NC + col0 + m];
    b.y = W[(long)(k0 + 1) * NC + col0 + m];
    // (neg_a, A, neg_b, B, c_mod, C, reuse_a, reuse_b)
    c = __builtin_amdgcn_wmma_f32_16x16x4_f32(false, a, false, b,
                                              (short)0, c, false, false);
  }
  float* __restrict__ orow = Out + (long)row0 * NC + col0 + m;
#pragma unroll
  for (int j = 0; j < 8; ++j) {
    orow[(long)(j + 8 * half) * NC] = c[j];
  }
}

// ---------------------------------------------------------------------------
// edge helpers (edge_index is [2, E] row-major int32; self-loops appended)
// ---------------------------------------------------------------------------
__device__ __forceinline__ int edge_src(const int* __restrict__ ei, long e) {
  return (e < NEDGES) ? ei[e] : (int)(e - NEDGES);
}
__device__ __forceinline__ int edge_dst(const int* __restrict__ ei, long e) {
  return (e < NEDGES) ? ei[NEDGES + e] : (int)(e - NEDGES);
}

// monotonic-bits fp32 atomic max (dest pre-initialized to -inf)
__device__ __forceinline__ void atomic_max_f32(float* addr, float val) {
  if (val >= 0.0f) {
    atomicMax((int*)addr, __float_as_int(val));
  } else {
    atomicMin((unsigned int*)addr, __float_as_uint(val));
  }
}

// ---------------------------------------------------------------------------
// Pass A: per (edge, head) attention logit + segment-max into mmax[dst,h]
// ---------------------------------------------------------------------------
template <int HEADS, int CH>
__global__ void gat_edge_logits(const int* __restrict__ ei,
                                const float* __restrict__ xl,
                                const float* __restrict__ xr,
                                const float* __restrict__ att,
                                float* __restrict__ logits,
                                float* __restrict__ mmax) {
  const long idx = (long)blockIdx.x * blockDim.x + threadIdx.x;
  const long tot = (long)ETOT * HEADS;
  if (idx >= tot) return;
  const long e = idx / HEADS;
  const int  h = (int)(idx % HEADS);
  const int  s = edge_src(ei, e);
  const int  d = edge_dst(ei, e);
  const float* __restrict__ pl = xl + (long)s * (HEADS * CH) + h * CH;
  const float* __restrict__ pr = xr + (long)d * (HEADS * CH) + h * CH;
  const float* __restrict__ pa = att + h * CH;
  float acc = 0.0f;
#pragma unroll
  for (int c = 0; c < CH; ++c) {
    float v = pl[c] + pr[c];
    v = (v > 0.0f) ? v : NEG_SLOPE * v;
    acc = fmaf(pa[c], v, acc);
  }
  logits[idx] = acc;
  atomic_max_f32(&mmax[(long)d * HEADS + h], acc);
}

// ---------------------------------------------------------------------------
// Pass B: a = exp(logit - max); segment-sum into denom[dst,h]
// ---------------------------------------------------------------------------
template <int HEADS>
__global__ void gat_edge_exp(const int* __restrict__ ei,
                             float* __restrict__ logits,
                             const float* __restrict__ mmax,
                             float* __restrict__ denom) {
  const long idx = (long)blockIdx.x * blockDim.x + threadIdx.x;
  const long tot = (long)ETOT * HEADS;
  if (idx >= tot) return;
  const long e = idx / HEADS;
  const int  h = (int)(idx % HEADS);
  const int  d = edge_dst(ei, e);
  const float a = __expf(logits[idx] - mmax[(long)d * HEADS + h]);
  logits[idx] = a;  // reuse buffer to hold exp values
  atomicAdd(&denom[(long)d * HEADS + h], a);
}

// ---------------------------------------------------------------------------
// Pass C, layer 1 (HC = 128): ONE WAVE32 PER EDGE.
// Edge id is wave-uniform (readfirstlane) so the src/dst index fetches become
// scalar SMEM loads; each lane gathers 4 consecutive features with a single
// b128 load and issues 4 f32 atomics into the L2-resident accumulator.
// ---------------------------------------------------------------------------
template <int HEADS, int CH>   // requires HEADS*CH == 128, CH % 4 == 0
__global__ void gat_edge_aggr_wave(const int* __restrict__ ei,
                                   const float* __restrict__ xl,
                                   const float* __restrict__ aexp,
                                   const float* __restrict__ denom,
                                   float* __restrict__ accum) {
  const int  wpb  = blockDim.x >> 5;
  const int  wid  = __builtin_amdgcn_readfirstlane((int)(threadIdx.x >> 5));
  const long e    = (long)blockIdx.x * wpb + wid;            // wave-uniform
  if (e >= ETOT) return;
  const int  lane = threadIdx.x & 31;
  const int  s    = edge_src(ei, e);   // scalar load
  const int  d    = edge_dst(ei, e);   // scalar load
  const int  f0   = lane * 4;          // features [f0, f0+4)
  const int  h    = f0 / CH;
  const float alpha = aexp[e * HEADS + h] / denom[(long)d * HEADS + h];
  const float4 v = *(const float4*)(xl + (long)s * (HEADS * CH) + f0);
  float* dst = accum + (long)d * (HEADS * CH) + f0;
  atomicAdd(dst + 0, v.x * alpha);
  atomicAdd(dst + 1, v.y * alpha);
  atomicAdd(dst + 2, v.z * alpha);
  atomicAdd(dst + 3, v.w * alpha);
}

// ---------------------------------------------------------------------------
// Pass C, generic thread-per-(edge, feature) path (used for layer 2, HC=16)
// ---------------------------------------------------------------------------
template <int HEADS, int CH>
__global__ void gat_edge_aggr(const int* __restrict__ ei,
                              const float* __restrict__ xl,
                              const float* __restrict__ aexp,
                              const float* __restrict__ denom,
                              float* __restrict__ accum) {
  const long idx = (long)blockIdx.x * blockDim.x + threadIdx.x;
  const long tot = (long)ETOT * HEADS * CH;
  if (idx >= tot) return;
  const int  hc = HEADS * CH;
  const long e  = idx / hc;
  const int  f  = (int)(idx % hc);
  const int  h  = f / CH;
  const int  s  = edge_src(ei, e);
  const int  d  = edge_dst(ei, e);
  const float alpha = aexp[e * HEADS + h] / denom[(long)d * HEADS + h];
  atomicAdd(&accum[(long)d * hc + f], xl[(long)s * hc + f] * alpha);
}

// ---------------------------------------------------------------------------
// Layer-1 epilogue: h = elu(accum + bias), vectorized x4 (HC1 % 4 == 0)
// ---------------------------------------------------------------------------
__global__ void gat_bias_elu4(const float* __restrict__ accum,
                              const float* __restrict__ bias,
                              float* __restrict__ hout, long n4, int hc) {
  const long i = (long)blockIdx.x * blockDim.x + threadIdx.x;
  if (i >= n4) return;
  const int f0 = (int)((i * 4) % hc);
  float4 v = *(const float4*)(accum + i * 4);
  const float4 b = *(const float4*)(bias + f0);
  v.x += b.x; v.y += b.y; v.z += b.z; v.w += b.w;
  v.x = (v.x > 0.0f) ? v.x : (__expf(v.x) - 1.0f);
  v.y = (v.y > 0.0f) ? v.y : (__expf(v.y) - 1.0f);
  v.z = (v.z > 0.0f) ? v.z : (__expf(v.z) - 1.0f);
  v.w = (v.w > 0.0f) ? v.w : (__expf(v.w) - 1.0f);
  *(float4*)(hout + i * 4) = v;
}

// ---------------------------------------------------------------------------
// Layer-2 epilogue: y = accum + bias2 -> d_out[0:N*16];
// log_softmax(y) -> d_out[N*16 : 2*N*16]
// ---------------------------------------------------------------------------
__global__ void gat_finalize(const float* __restrict__ accum,
                             const float* __restrict__ bias,
                             float* __restrict__ out) {
  const int nidx = blockIdx.x * blockDim.x + threadIdx.x;
  if (nidx >= NNODES) return;
  float y[C2];
  float mx = -__builtin_huge_valf();
#pragma unroll
  for (int f = 0; f < C2; ++f) {
    y[f] = accum[(long)nidx * C2 + f] + bias[f];
    mx = fmaxf(mx, y[f]);
  }
  float se = 0.0f;
#pragma unroll
  for (int f = 0; f < C2; ++f) se += __expf(y[f] - mx);
  const float lse = mx + __logf(se);
#pragma unroll
  for (int f = 0; f < C2; ++f) {
    out[(long)nidx * C2 + f] = y[f];
    out[(long)NNODES * C2 + (long)nidx * C2 + f] = y[f] - lse;
  }
}

// ---------------------------------------------------------------------------
// host launch
// ---------------------------------------------------------------------------
static inline unsigned blks(long n, int bs) { return (unsigned)((n + bs - 1) / bs); }

extern "C" void kernel_launch(void* const* d_in, const int* in_sizes, int n_in,
                              void* d_out, int out_size, void* d_ws, size_t ws_size,
                              hipStream_t stream) {
  (void)in_sizes; (void)n_in; (void)out_size; (void)ws_size;
  const float* x     = (const float*)d_in[0];
  const int*   ei    = (const int*)  d_in[1];
  const float* Wl1   = (const float*)d_in[2];
  const float* Wr1   = (const float*)d_in[3];
  const float* att1  = (const float*)d_in[4];
  const float* bias1 = (const float*)d_in[5];
  const float* Wl2   = (const float*)d_in[6];
  const float* Wr2   = (const float*)d_in[7];
  const float* att2  = (const float*)d_in[8];
  const float* bias2 = (const float*)d_in[9];
  float* out = (float*)d_out;

  // workspace carve (floats)
  float* ws = (float*)d_ws;
  long off = 0;
  float* xl1    = ws + off; off += (long)NNODES * HC1;
  float* xr1    = ws + off; off += (long)NNODES * HC1;
  float* logits = ws + off; off += (long)ETOT * H1;      // reused as exp vals
  float* m1     = ws + off; off += (long)NNODES * H1;
  float* den1   = ws + off; off += (long)NNODES * H1;
  float* acc1   = ws + off; off += (long)NNODES * HC1;
  float* hbuf   = ws + off; off += (long)NNODES * HC1;
  float* xl2    = ws + off; off += (long)NNODES * C2;
  float* xr2    = ws + off; off += (long)NNODES * C2;
  float* logit2 = ws + off; off += (long)ETOT;
  float* m2     = ws + off; off += (long)NNODES;
  float* den2   = ws + off; off += (long)NNODES;
  float* acc2   = ws + off; off += (long)NNODES * C2;

  const int BS = 256;
  const float NEG_INF = -__builtin_huge_valf();

  // init accumulators / softmax state every call (ws is not re-poisoned)
  gat_fill<<<blks((long)NNODES * H1, BS),  BS, 0, stream>>>(m1,  NEG_INF, (long)NNODES * H1);
  gat_fill<<<blks((long)NNODES * H1, BS),  BS, 0, stream>>>(den1, 0.0f,   (long)NNODES * H1);
  gat_fill<<<blks((long)NNODES * HC1, BS), BS, 0, stream>>>(acc1, 0.0f,   (long)NNODES * HC1);
  gat_fill<<<blks((long)NNODES, BS),       BS, 0, stream>>>(m2,  NEG_INF, (long)NNODES);
  gat_fill<<<blks((long)NNODES, BS),       BS, 0, stream>>>(den2, 0.0f,   (long)NNODES);
  gat_fill<<<blks((long)NNODES * C2, BS),  BS, 0, stream>>>(acc2, 0.0f,   (long)NNODES * C2);

  // ---- layer 1 ----
  {
    const long waves = ((long)NNODES / 16) * (HC1 / 16);   // 25000 tiles
    const unsigned g = blks(waves * 32, BS);
    gat_wmma_gemm_f32<<<g, BS, 0, stream>>>(x, Wl1, xl1, NNODES, DIN, HC1);
    gat_wmma_gemm_f32<<<g, BS, 0, stream>>>(x, Wr1, xr1, NNODES, DIN, HC1);
  }
  gat_edge_logits<H1, C1><<<blks((long)ETOT * H1, BS), BS, 0, stream>>>(
      ei, xl1, xr1, att1, logits, m1);
  gat_edge_exp<H1><<<blks((long)ETOT * H1, BS), BS, 0, stream>>>(
      ei, logits, m1, den1);
  {
    const int wpb = BS / 32;                               // 8 edges per block
    gat_edge_aggr_wave<H1, C1><<<blks((long)ETOT, wpb), BS, 0, stream>>>(
        ei, xl1, logits, den1, acc1);
  }
  gat_bias_elu4<<<blks((long)NNODES * HC1 / 4, BS), BS, 0, stream>>>(
      acc1, bias1, hbuf, (long)NNODES * HC1 / 4, HC1);

  // ---- layer 2 ----
  {
    const long waves = ((long)NNODES / 16) * (C2 / 16);    // 3125 tiles
    const unsigned g = blks(waves * 32, BS);
    gat_wmma_gemm_f32<<<g, BS, 0, stream>>>(hbuf, Wl2, xl2, NNODES, HC1, C2);
    gat_wmma_gemm_f32<<<g, BS, 0, stream>>>(hbuf, Wr2, xr2, NNODES, HC1, C2);
  }
  gat_edge_logits<H2, C2><<<blks((long)ETOT, BS), BS, 0, stream>>>(
      ei, xl2, xr2, att2, logit2, m2);
  gat_edge_exp<H2><<<blks((long)ETOT, BS), BS, 0, stream>>>(
      ei, logit2, m2, den2);
  gat_edge_aggr<H2, C2><<<blks((long)ETOT * C2, BS), BS, 0, stream>>>(
      ei, xl2, logit2, den2, acc2);
  gat_finalize<<<blks((long)NNODES, BS), BS, 0, stream>>>(acc2, bias2, out);
}